// IntraInterLoss_7541962572442
// MI455X (gfx1250) — compile-verified
//
#include <hip/hip_runtime.h>

typedef float v2f __attribute__((ext_vector_type(2)));
typedef float v8f __attribute__((ext_vector_type(8)));
typedef int   v4i __attribute__((ext_vector_type(4)));

#define NROWS 65536
#define DCOLS 768
#define RCHUNK 16            // rows staged per chunk
#define LDS_STRIDE 784       // 768 + 16 pad floats -> breaks LDS bank conflicts
#define NBLOCKS 512
#define NTHREADS 256         // 8 waves (wave32)

// Async global->LDS path (CDNA5): copy bypasses VGPRs, tracked by ASYNCcnt.
#if defined(__has_builtin)
#if __has_builtin(__builtin_amdgcn_global_load_async_to_lds_b128) && \
    __has_builtin(__builtin_amdgcn_s_wait_asynccnt)
#define USE_ASYNC_LDS 1
#endif
#endif
#ifndef USE_ASYNC_LDS
#define USE_ASYNC_LDS 0
#endif

#if USE_ASYNC_LDS
typedef __attribute__((address_space(1))) v4i* gv4i_ptr;   // global (device) int4*
typedef __attribute__((address_space(3))) v4i* lv4i_ptr;   // LDS int4*
#endif

// ---------------- zero workspace ----------------
__global__ void zero_ws_kernel(float* ws, int n) {
    int i = blockIdx.x * blockDim.x + threadIdx.x;
    if (i < n) ws[i] = 0.0f;
}

// ---------------- main streaming + WMMA kernel ----------------
// ws layout: [0..767] = s0 accum, [768..1535] = s1 accum, word 1536 = count(target==1)
__global__ __launch_bounds__(NTHREADS)
void masked_norm_sum_kernel(const float* __restrict__ emb,
                            const int* __restrict__ target,
                            float* __restrict__ ws) {
    extern __shared__ float smem[];
    float* sdata = smem;                       // RCHUNK x LDS_STRIDE
    float* w0 = smem + RCHUNK * LDS_STRIDE;    // [RCHUNK]
    float* w1 = w0 + RCHUNK;                   // [RCHUNK]

    const int lane = threadIdx.x & 31;
    const int wave = threadIdx.x >> 5;
    const int m    = lane & 15;                // M for A-frag / N for B-frag
    const int half = lane >> 4;

    v8f acc[6];
    #pragma unroll
    for (int t = 0; t < 6; ++t) acc[t] = (v8f){0,0,0,0,0,0,0,0};

    int cnt1 = 0;
    const int nchunks = NROWS / RCHUNK;        // 4096

    for (int chunk = blockIdx.x; chunk < nchunks; chunk += NBLOCKS) {
        // -------- Phase A: stage 16 rows into LDS, compute rinv per row --------
#if USE_ASYNC_LDS
        // Issue all async global->LDS copies for this wave's 2 rows (no VGPR staging).
        #pragma unroll
        for (int rr = 0; rr < 2; ++rr) {
            const int  r = wave * 2 + rr;
            const long g = (long)chunk * RCHUNK + r;
            const float* gsrc  = emb + g * (long)DCOLS;
            float*       lrow  = sdata + r * LDS_STRIDE;

            const long gn = g + (long)NBLOCKS * RCHUNK;
            if (gn < NROWS)
                __builtin_prefetch(emb + gn * (long)DCOLS + lane * 4, 0, 1);

            #pragma unroll
            for (int p = 0; p < 6; ++p) {
                const int off = 4 * (lane + 32 * p);         // float offset
                __builtin_amdgcn_global_load_async_to_lds_b128(
                    (gv4i_ptr)(gsrc + off),
                    (lv4i_ptr)(lrow + off),
                    0, 0);
            }
        }
        __builtin_amdgcn_s_wait_asynccnt(0);   // our copies are now in LDS

        // Row sum-of-squares from the staged LDS data.
        #pragma unroll
        for (int rr = 0; rr < 2; ++rr) {
            const int  r = wave * 2 + rr;
            const long g = (long)chunk * RCHUNK + r;
            const float4* lrow4 = (const float4*)(sdata + r * LDS_STRIDE);
            float ss = 0.0f;
            #pragma unroll
            for (int p = 0; p < 6; ++p) {
                float4 v = lrow4[lane + 32 * p];
                ss += v.x * v.x + v.y * v.y + v.z * v.z + v.w * v.w;
            }
            #pragma unroll
            for (int sh = 16; sh >= 1; sh >>= 1)
                ss += __shfl_xor(ss, sh, 32);
            const float rinv = rsqrtf(ss);
            if (lane == 0) {
                const int tv = target[g];
                w0[r] = (tv == 0) ? rinv : 0.0f;
                w1[r] = (tv != 0) ? rinv : 0.0f;
                cnt1 += (tv != 0);
            }
        }
#else
        #pragma unroll
        for (int rr = 0; rr < 2; ++rr) {
            const int  r = wave * 2 + rr;                    // row within chunk
            const long g = (long)chunk * RCHUNK + r;         // global row
            const float4* src = (const float4*)(emb + g * (long)DCOLS);

            const long gn = g + (long)NBLOCKS * RCHUNK;
            if (gn < NROWS)
                __builtin_prefetch(emb + gn * (long)DCOLS + lane * 4, 0, 1);

            float4 vals[6];
            float  ss = 0.0f;
            #pragma unroll
            for (int p = 0; p < 6; ++p) {                    // 24 floats/lane
                float4 v = src[lane + 32 * p];
                vals[p] = v;
                ss += v.x * v.x + v.y * v.y + v.z * v.z + v.w * v.w;
            }
            #pragma unroll
            for (int sh = 16; sh >= 1; sh >>= 1)
                ss += __shfl_xor(ss, sh, 32);
            const float rinv = rsqrtf(ss);

            float4* dst = (float4*)(sdata + r * LDS_STRIDE);
            #pragma unroll
            for (int p = 0; p < 6; ++p) dst[lane + 32 * p] = vals[p];

            if (lane == 0) {
                const int tv = target[g];
                w0[r] = (tv == 0) ? rinv : 0.0f;
                w1[r] = (tv != 0) ? rinv : 0.0f;
                cnt1 += (tv != 0);
            }
        }
#endif
        __syncthreads();

        // -------- Phase B: S[2x768] += W[2x16] * X[16x768] via WMMA f32 --------
        // A (16x4 f32): M = lane&15, K = vgpr + 2*(lane>>4)
        // B (4x16 f32): N = lane&15, K = vgpr + 2*(lane>>4)
        #pragma unroll
        for (int kb = 0; kb < RCHUNK / 4; ++kb) {
            const int kk = kb * 4 + 2 * half;                // K row for vgpr0
            v2f a; a.x = 0.0f; a.y = 0.0f;
            if (m == 0)      { a.x = w0[kk]; a.y = w0[kk + 1]; }
            else if (m == 1) { a.x = w1[kk]; a.y = w1[kk + 1]; }

            #pragma unroll
            for (int t = 0; t < 6; ++t) {
                const int col = wave * 96 + t * 16 + m;      // global column
                v2f b;
                b.x = sdata[kk * LDS_STRIDE + col];
                b.y = sdata[(kk + 1) * LDS_STRIDE + col];
                acc[t] = __builtin_amdgcn_wmma_f32_16x16x4_f32(
                    false, a, false, b, (short)0, acc[t], false, false);
            }
        }
        __syncthreads();   // protect LDS before next chunk's Phase A
    }

    // -------- Epilogue: C rows M=0 (s0) and M=1 (s1) live in vgpr0/1, lanes 0-15 --------
    if (lane < 16) {
        #pragma unroll
        for (int t = 0; t < 6; ++t) {
            const int col = wave * 96 + t * 16 + lane;
            atomicAdd(&ws[col],         acc[t][0]);   // s0
            atomicAdd(&ws[DCOLS + col], acc[t][1]);   // s1
        }
    }
    if (lane == 0 && cnt1 > 0)
        atomicAdd((unsigned int*)(ws + 2 * DCOLS), (unsigned int)cnt1);
}

// ---------------- final tiny reduction ----------------
__global__ void finalize_kernel(const float* __restrict__ ws, float* __restrict__ out) {
    const float* s0 = ws;
    const float* s1 = ws + DCOLS;
    float p00 = 0.0f, p11 = 0.0f, p01 = 0.0f;
    for (int col = threadIdx.x; col < DCOLS; col += 256) {
        const float x0 = s0[col], x1 = s1[col];
        p00 += x0 * x0; p11 += x1 * x1; p01 += x0 * x1;
    }
    #pragma unroll
    for (int sh = 16; sh >= 1; sh >>= 1) {
        p00 += __shfl_xor(p00, sh, 32);
        p11 += __shfl_xor(p11, sh, 32);
        p01 += __shfl_xor(p01, sh, 32);
    }
    __shared__ float r00[8], r11[8], r01[8];
    const int lane = threadIdx.x & 31, wave = threadIdx.x >> 5;
    if (lane == 0) { r00[wave] = p00; r11[wave] = p11; r01[wave] = p01; }
    __syncthreads();
    if (threadIdx.x == 0) {
        float t00 = 0.0f, t11 = 0.0f, t01 = 0.0f;
        for (int w = 0; w < 8; ++w) { t00 += r00[w]; t11 += r11[w]; t01 += r01[w]; }
        const unsigned c1u = *(const unsigned int*)(ws + 2 * DCOLS);
        const float c1 = (float)c1u;
        const float c0 = (float)NROWS - c1;
        // r_nr_cos - 0.4*r_cos - 0.1*nr_cos
        out[0] = t01 / (c1 * c0) - 0.4f * t11 / (c1 * c1) - 0.1f * t00 / (c0 * c0);
    }
}

extern "C" void kernel_launch(void* const* d_in, const int* in_sizes, int n_in,
                              void* d_out, int out_size, void* d_ws, size_t ws_size,
                              hipStream_t stream) {
    (void)in_sizes; (void)n_in; (void)out_size; (void)ws_size;
    const float* emb    = (const float*)d_in[0];
    const int*   target = (const int*)d_in[1];
    float*       ws     = (float*)d_ws;

    const int ws_words = 2 * DCOLS + 1;                        // s0, s1, count
    zero_ws_kernel<<<(ws_words + 255) / 256, 256, 0, stream>>>(ws, ws_words);

    const size_t smem_bytes = (RCHUNK * LDS_STRIDE + 2 * RCHUNK) * sizeof(float); // ~50 KB
    masked_norm_sum_kernel<<<NBLOCKS, NTHREADS, smem_bytes, stream>>>(emb, target, ws);

    finalize_kernel<<<1, 256, 0, stream>>>(ws, (float*)d_out);
}